// TabNetEncoder_41042707481072
// MI455X (gfx1250) — compile-verified
//
#include <hip/hip_runtime.h>
#include <cstddef>
#include <cstdint>

// ---------------- constants ----------------
#define BATCH   32768
#define INP     512
#define ND      64
#define NA      64
#define NSTEPS  3
#define VBS     512
#define MDIM    128
#define H2      256          // 2*MDIM
#define NCHUNK  (BATCH / VBS)   // 64
#define GAMMA_C 1.3f
#define EPS_C   1e-15f
#define BNEPS_C 1e-5f
#define SQRTH_C 0.70710678118654752440f

typedef __attribute__((ext_vector_type(2))) float v2f;
typedef __attribute__((ext_vector_type(4))) float v4f;
typedef __attribute__((ext_vector_type(8))) float v8f;

// =====================================================================
// WMMA fp32 GEMM:  C[M x N] = A[M x K](row-major, leading dim lda) @ W[K x N]
// Block tile 64x64, 8 waves, each wave: one 16-row tile x two 16-col tiles.
// Staging: double-buffered LDS tiles filled with GLOBAL_LOAD_ASYNC_TO_LDS_B128
// (ASYNCcnt-tracked, no VGPR bounce, next tile copies while WMMAs run).
// =====================================================================
#define BM 64
#define BN 64
#define BK 16
#define APAD 20   // padded LDS row (multiple of 4 for 16B-aligned b128 writes)

__device__ __forceinline__ void async_copy_b128(uint32_t lds_addr, const float* gptr)
{
    asm volatile("global_load_async_to_lds_b128 %0, %1, off"
                 :: "v"(lds_addr), "v"(gptr) : "memory");
}

__global__ __launch_bounds__(256)
void wmma_gemm_f32(const float* __restrict__ A, const float* __restrict__ W,
                   float* __restrict__ C, int M, int N, int K, int lda)
{
    __shared__ float As[2][BM][APAD];
    __shared__ float Bs[2][BK][BN];

    const int tid  = threadIdx.x;
    const int wave = tid >> 5;          // 0..7
    const int lane = tid & 31;
    const int m0 = blockIdx.y * BM;
    const int n0 = blockIdx.x * BN;

    const int wm = (wave & 3) * 16;     // wave row offset within tile
    const int wn = (wave >> 2) * 32;    // wave col offset (two 16-wide tiles)

    const int lm = lane & 15;           // A-frag row / B-frag col
    const int kh = lane >> 4;           // K half-select (0/1)

    // per-thread staging coordinates (one b128 each for A and B tile)
    const int ar = tid >> 2;            // 0..63
    const int ac = (tid & 3) * 4;       // 0,4,8,12
    const int br = tid >> 4;            // 0..15
    const int bc = (tid & 15) * 4;      // 0..60

    v8f acc0 = {};
    v8f acc1 = {};

    const int nst = K / BK;

    auto issue_stage = [&](int k0, int bf) {
        uint32_t la = (uint32_t)(uintptr_t)&As[bf][ar][ac];
        async_copy_b128(la, A + (size_t)(m0 + ar) * lda + k0 + ac);
        uint32_t lb = (uint32_t)(uintptr_t)&Bs[bf][br][bc];
        async_copy_b128(lb, W + (size_t)(k0 + br) * N + n0 + bc);
    };

    int buf = 0;
    issue_stage(0, 0);

    for (int i = 0; i < nst; ++i) {
        if (i + 1 < nst) {
            issue_stage((i + 1) * BK, buf ^ 1);
            // older pair (this stage) done; keep the 2 just-issued in flight
            asm volatile("s_wait_asynccnt 0x2" ::: "memory");
        } else {
            asm volatile("s_wait_asynccnt 0x0" ::: "memory");
        }
        __syncthreads();   // all waves' copies for `buf` landed

#pragma unroll
        for (int kk = 0; kk < BK; kk += 4) {
            v2f a;
            a.x = As[buf][wm + lm][kk + 2 * kh + 0];
            a.y = As[buf][wm + lm][kk + 2 * kh + 1];
            v2f b0, b1;
            b0.x = Bs[buf][kk + 2 * kh + 0][wn + lm];
            b0.y = Bs[buf][kk + 2 * kh + 1][wn + lm];
            b1.x = Bs[buf][kk + 2 * kh + 0][wn + 16 + lm];
            b1.y = Bs[buf][kk + 2 * kh + 1][wn + 16 + lm];
            acc0 = __builtin_amdgcn_wmma_f32_16x16x4_f32(false, a, false, b0,
                                                         (short)0, acc0, false, false);
            acc1 = __builtin_amdgcn_wmma_f32_16x16x4_f32(false, a, false, b1,
                                                         (short)0, acc1, false, false);
        }
        __syncthreads();   // everyone done reading `buf` before it is refilled
        buf ^= 1;
    }

    // store: lane L -> n = L&15 ; m = (L>>4)*8 + v
    const int cn = lane & 15;
    const int cm = (lane >> 4) * 8;
#pragma unroll
    for (int v = 0; v < 8; ++v) {
        C[(size_t)(m0 + wm + cm + v) * N + n0 + wn + cn]      = acc0[v];
        C[(size_t)(m0 + wm + cm + v) * N + n0 + wn + 16 + cn] = acc1[v];
    }
}

// =====================================================================
// Ghost-BN chunk stats: per 512-row chunk, per column mean & rstd
// =====================================================================
__global__ __launch_bounds__(256)
void gbn_chunk_stats(const float* __restrict__ Y, int ncols,
                     float* __restrict__ mean, float* __restrict__ rstd)
{
    const int chunk = blockIdx.x;
    const float* base = Y + (size_t)chunk * VBS * ncols;
    for (int col = threadIdx.x; col < ncols; col += 256) {
        float s = 0.f, q = 0.f;
        for (int r = 0; r < VBS; ++r) {
            float v = base[(size_t)r * ncols + col];
            s += v; q += v * v;
        }
        float mu  = s * (1.0f / VBS);
        float var = q * (1.0f / VBS) - mu * mu;
        mean[chunk * ncols + col] = mu;
        rstd[chunk * ncols + col] = rsqrtf(var + BNEPS_C);
    }
}

// =====================================================================
// GLU combine: x_out = (x_prev? + bn(Y[:, :128]) * sigmoid(bn(Y[:,128:]))) * s
// optional relu(first 64 cols) -> step_out (d_out slice)
// =====================================================================
__global__ __launch_bounds__(256)
void glu_combine(const float* __restrict__ Y,
                 const float* __restrict__ mean, const float* __restrict__ rstd,
                 const float* __restrict__ g, const float* __restrict__ b,
                 const float* __restrict__ xprev, float* __restrict__ xout,
                 float* __restrict__ step_out, int residual)
{
    int idx = blockIdx.x * 256 + threadIdx.x;      // over BATCH*128
    if (idx >= BATCH * MDIM) return;
    int row = idx >> 7;
    int col = idx & (MDIM - 1);
    int chunk = row >> 9;
    const float* mu = mean + chunk * H2;
    const float* rs = rstd + chunk * H2;
    size_t yb = (size_t)row * H2;

    float hv = (Y[yb + col] - mu[col]) * rs[col] * g[col] + b[col];
    int cg = col + MDIM;
    float hg = (Y[yb + cg] - mu[cg]) * rs[cg] * g[cg] + b[cg];
    float glu = hv * (1.0f / (1.0f + __expf(-hg)));
    float out = residual ? (xprev[idx] + glu) * SQRTH_C : glu;
    xout[idx] = out;
    if (step_out != nullptr && col < ND)
        step_out[(size_t)row * ND + col] = fmaxf(out, 0.0f);
}

// =====================================================================
// bn0 (full-batch BN) pipeline
// =====================================================================
__global__ __launch_bounds__(256)
void bn0_partial(const float* __restrict__ x,
                 float* __restrict__ ps, float* __restrict__ pq)
{
    const int c = blockIdx.x;                        // 64 row-chunks
    const float* base = x + (size_t)c * VBS * INP;
    for (int col = threadIdx.x; col < INP; col += 256) {
        float s = 0.f, q = 0.f;
        for (int r = 0; r < VBS; ++r) {
            float v = base[(size_t)r * INP + col];
            s += v; q += v * v;
        }
        ps[c * INP + col] = s;
        pq[c * INP + col] = q;
    }
}

__global__ __launch_bounds__(256)
void bn0_finalize(const float* __restrict__ ps, const float* __restrict__ pq,
                  float* __restrict__ bmu, float* __restrict__ brs)
{
    int col = blockIdx.x * 256 + threadIdx.x;
    if (col >= INP) return;
    float s = 0.f, q = 0.f;
    for (int c = 0; c < NCHUNK; ++c) { s += ps[c * INP + col]; q += pq[c * INP + col]; }
    float mu  = s * (1.0f / BATCH);
    float var = q * (1.0f / BATCH) - mu * mu;
    bmu[col] = mu;
    brs[col] = rsqrtf(var + BNEPS_C);
}

__global__ __launch_bounds__(256)
void bn0_apply(const float* __restrict__ x, const float* __restrict__ bmu,
               const float* __restrict__ brs, const float* __restrict__ g,
               const float* __restrict__ b, float* __restrict__ xn)
{
    int idx = blockIdx.x * 256 + threadIdx.x;
    if (idx >= BATCH * INP) return;
    int col = idx & (INP - 1);
    xn[idx] = (x[idx] - bmu[col]) * brs[col] * g[col] + b[col];
}

// =====================================================================
// attention: a = gbn(att@W) * prior   (in place over the GEMM output)
// =====================================================================
__global__ __launch_bounds__(256)
void att_bn_mask(float* __restrict__ Y,
                 const float* __restrict__ mean, const float* __restrict__ rstd,
                 const float* __restrict__ g, const float* __restrict__ b,
                 const float* __restrict__ prior)
{
    int idx = blockIdx.x * 256 + threadIdx.x;
    if (idx >= BATCH * INP) return;
    int row = idx >> 9;
    int col = idx & (INP - 1);
    int chunk = row >> 9;
    float h = (Y[idx] - mean[chunk * INP + col]) * rstd[chunk * INP + col] * g[col] + b[col];
    Y[idx] = h * prior[idx];
}

// =====================================================================
// Fused sparsemax per row (512 elems, 256 threads):
//   M = sparsemax(a); mloss += sum(M*log(M+eps)); prior=(g-M)*prior; Mx=M*xn
// =====================================================================
__global__ __launch_bounds__(256)
void sparsemax_fused(const float* __restrict__ a, float* __restrict__ prior,
                     const float* __restrict__ xn, float* __restrict__ Mx,
                     float* __restrict__ mloss)
{
    __shared__ float z[INP];
    __shared__ float srt[INP];
    __shared__ float cs[INP];
    __shared__ float red[256];

    const int row = blockIdx.x;
    const int t = threadIdx.x;
    const float* arow = a + (size_t)row * INP;

    float v0 = arow[t];
    float v1 = arow[t + 256];

    // row max
    red[t] = fmaxf(v0, v1);
    __syncthreads();
    for (int s = 128; s > 0; s >>= 1) {
        if (t < s) red[t] = fmaxf(red[t], red[t + s]);
        __syncthreads();
    }
    float rowmax = red[0];
    __syncthreads();

    z[t] = v0 - rowmax;  z[t + 256] = v1 - rowmax;
    srt[t] = z[t];       srt[t + 256] = z[t + 256];

    // bitonic sort descending (512 elems)
    for (unsigned size = 2; size <= INP; size <<= 1) {
        for (unsigned stride = size >> 1; stride > 0; stride >>= 1) {
            __syncthreads();
            unsigned i = (unsigned)t;
            unsigned idx = (i / stride) * (stride << 1) + (i % stride);
            unsigned partner = idx + stride;
            bool desc = ((idx & size) == 0);
            float x0 = srt[idx], x1 = srt[partner];
            if ((x0 < x1) == desc) { srt[idx] = x1; srt[partner] = x0; }
        }
    }
    __syncthreads();

    // inclusive cumsum (Hillis-Steele, 2 elems/thread)
    cs[t] = srt[t];  cs[t + 256] = srt[t + 256];
    __syncthreads();
    for (int off = 1; off < INP; off <<= 1) {
        float a0 = cs[t],       b0 = (t >= off) ? cs[t - off] : 0.f;
        float a1 = cs[t + 256], b1 = ((t + 256) >= off) ? cs[t + 256 - off] : 0.f;
        __syncthreads();
        cs[t] = a0 + b0;  cs[t + 256] = a1 + b1;
        __syncthreads();
    }

    // k = #{ rho * z_srt > cumsum - 1 }
    int cnt = 0;
    if ((float)(t + 1)   * srt[t]       > cs[t]       - 1.0f) cnt++;
    if ((float)(t + 257) * srt[t + 256] > cs[t + 256] - 1.0f) cnt++;
    red[t] = (float)cnt;
    __syncthreads();
    for (int s = 128; s > 0; s >>= 1) {
        if (t < s) red[t] += red[t + s];
        __syncthreads();
    }
    int k = (int)red[0];
    float tau = (cs[k - 1] - 1.0f) / (float)k;
    __syncthreads();

    // emit M-derived quantities
    float ml = 0.f;
    for (int e = t; e < INP; e += 256) {
        float Mv = fmaxf(z[e] - tau, 0.f);
        ml += Mv * logf(Mv + EPS_C);
        size_t gi = (size_t)row * INP + e;
        float pr = prior[gi];
        prior[gi] = (GAMMA_C - Mv) * pr;
        Mx[gi] = Mv * xn[gi];
    }
    red[t] = ml;
    __syncthreads();
    for (int s = 128; s > 0; s >>= 1) {
        if (t < s) red[t] += red[t + s];
        __syncthreads();
    }
    if (t == 0) atomicAdd(mloss, red[0]);
}

__global__ void init_prior(float* __restrict__ prior, float* __restrict__ acc)
{
    int idx = blockIdx.x * 256 + threadIdx.x;
    if (idx < BATCH * INP) prior[idx] = 1.0f;
    if (idx == 0) acc[0] = 0.0f;
}

__global__ void finalize_mloss(const float* __restrict__ acc, float* __restrict__ out)
{
    out[0] = acc[0] / (float)(BATCH * NSTEPS);
}

// =====================================================================
// host side
// =====================================================================
extern "C" void kernel_launch(void* const* d_in, const int* in_sizes, int n_in,
                              void* d_out, int out_size, void* d_ws, size_t ws_size,
                              hipStream_t stream)
{
    (void)in_sizes; (void)n_in; (void)out_size; (void)ws_size;

    // ---- inputs (setup_inputs dict order) ----
    const float* x   = (const float*)d_in[0];
    const float* g0  = (const float*)d_in[1];
    const float* b0  = (const float*)d_in[2];
    const float* sw0 = (const float*)d_in[3];   // 512x256
    const float* sw1 = (const float*)d_in[4];   // 128x256
    // splitter s: base 5+10*s : indW0, indW1, bn_g[0..3], bn_b[0..3]
    // att s:      base 45+3*s : W(64x512), g(512), b(512)

    // ---- workspace layout (floats) ----
    float* w = (float*)d_ws;
    float* xn    = w;  w += (size_t)BATCH * INP;
    float* Mx    = w;  w += (size_t)BATCH * INP;
    float* Y     = w;  w += (size_t)BATCH * INP;    // gemm out (max 32768x512)
    float* prior = w;  w += (size_t)BATCH * INP;
    float* xcur  = w;  w += (size_t)BATCH * MDIM;
    float* mean  = w;  w += (size_t)NCHUNK * INP;
    float* rstd  = w;  w += (size_t)NCHUNK * INP;
    float* psum  = w;  w += (size_t)NCHUNK * INP;
    float* psumq = w;  w += (size_t)NCHUNK * INP;
    float* bmu   = w;  w += INP;
    float* brs   = w;  w += INP;
    float* acc   = w;  w += 1;

    float* out = (float*)d_out;

    const int EW = (BATCH * INP) / 256;   // elementwise grid (B*512)
    const int EG = (BATCH * MDIM) / 256;  // elementwise grid (B*128)

    // ---- init + bn0 ----
    init_prior<<<EW, 256, 0, stream>>>(prior, acc);
    bn0_partial<<<NCHUNK, 256, 0, stream>>>(x, psum, psumq);
    bn0_finalize<<<2, 256, 0, stream>>>(psum, psumq, bmu, brs);
    bn0_apply<<<EW, 256, 0, stream>>>(x, bmu, brs, g0, b0, xn);

    // one GLU layer: Y = A@W ; chunk stats ; combine
    auto glu = [&](const float* Ain, int K, int lda, const float* W,
                   const float* gg, const float* bb, int residual, float* step_out) {
        dim3 grid(H2 / BN, BATCH / BM);
        wmma_gemm_f32<<<grid, 256, 0, stream>>>(Ain, W, Y, BATCH, H2, K, lda);
        gbn_chunk_stats<<<NCHUNK, 256, 0, stream>>>(Y, H2, mean, rstd);
        glu_combine<<<EG, 256, 0, stream>>>(Y, mean, rstd, gg, bb,
                                            xcur, xcur, step_out, residual);
    };

    auto transformer = [&](const float* input, int splitter, float* step_out) {
        int base = 5 + 10 * splitter;
        const float* iW0 = (const float*)d_in[base + 0];
        const float* iW1 = (const float*)d_in[base + 1];
        const float* bg0 = (const float*)d_in[base + 2];
        const float* bg1 = (const float*)d_in[base + 3];
        const float* bg2 = (const float*)d_in[base + 4];
        const float* bg3 = (const float*)d_in[base + 5];
        const float* bb0 = (const float*)d_in[base + 6];
        const float* bb1 = (const float*)d_in[base + 7];
        const float* bb2 = (const float*)d_in[base + 8];
        const float* bb3 = (const float*)d_in[base + 9];
        glu(input, INP,  INP,  sw0, bg0, bb0, 0, nullptr);
        glu(xcur,  MDIM, MDIM, sw1, bg1, bb1, 1, nullptr);
        glu(xcur,  MDIM, MDIM, iW0, bg2, bb2, 1, nullptr);
        glu(xcur,  MDIM, MDIM, iW1, bg3, bb3, 1, step_out);
    };

    // pre-step transformer -> att lives in xcur[:, 64:]
    transformer(xn, 0, nullptr);

    for (int s = 0; s < NSTEPS; ++s) {
        int abase = 45 + 3 * s;
        const float* aW = (const float*)d_in[abase + 0];
        const float* ag = (const float*)d_in[abase + 1];
        const float* ab = (const float*)d_in[abase + 2];

        // a = gbn(att @ W) * prior ; att = xcur[:,64:] (lda=128)
        dim3 grid(INP / BN, BATCH / BM);
        wmma_gemm_f32<<<grid, 256, 0, stream>>>(xcur + ND, aW, Y, BATCH, INP, NA, MDIM);
        gbn_chunk_stats<<<NCHUNK, 256, 0, stream>>>(Y, INP, mean, rstd);
        att_bn_mask<<<EW, 256, 0, stream>>>(Y, mean, rstd, ag, ab, prior);

        // sparsemax: mask entropy, prior update, Mx = M * xn
        sparsemax_fused<<<BATCH, 256, 0, stream>>>(Y, prior, xn, Mx, acc);

        // decision transformer; relu(o[:, :64]) -> d_out step slice
        transformer(Mx, s + 1, out + (size_t)s * BATCH * ND);
    }

    finalize_mloss<<<1, 1, 0, stream>>>(acc, out + (size_t)NSTEPS * BATCH * ND);
}